// CausalSelfAttention_6949257085500
// MI455X (gfx1250) — compile-verified
//
#include <hip/hip_runtime.h>
#include <hip/hip_bf16.h>
#include <math.h>

typedef __attribute__((ext_vector_type(2))) float v2f;
typedef __attribute__((ext_vector_type(8))) float v8f;
typedef __attribute__((ext_vector_type(4))) int   v4i;

// ---------------------------------------------------------------------------
// gfx1250 async global->LDS copy (ASYNCcnt-tracked), with fallback.
// Builtin signature (from hipcc diagnostic): (v4i addrspace(1)*, v4i addrspace(3)*, Ii, Ii)
// ---------------------------------------------------------------------------
#if defined(__AMDGCN__) && __has_builtin(__builtin_amdgcn_global_load_async_to_lds_b128) && \
    __has_builtin(__builtin_amdgcn_s_wait_asynccnt)
#define ASYNC_LDS 1
#else
#define ASYNC_LDS 0
#endif

__device__ __forceinline__ void async_cp_b128(void* lds_dst, const void* gsrc) {
#if ASYNC_LDS
    __builtin_amdgcn_global_load_async_to_lds_b128(
        (__attribute__((address_space(1))) v4i*)(gsrc),
        (__attribute__((address_space(3))) v4i*)(lds_dst),
        /*offset=*/0, /*cpol=*/0);
#else
    *reinterpret_cast<float4*>(lds_dst) = *reinterpret_cast<const float4*>(gsrc);
#endif
}

__device__ __forceinline__ void async_wait0() {
#if ASYNC_LDS
    __builtin_amdgcn_s_wait_asynccnt(0);
#endif
}

// D = A(16x4) * B(4x16) + C, exact fp32 on the matrix pipe.
__device__ __forceinline__ v8f wmma4(v2f a, v2f b, v8f c) {
    return __builtin_amdgcn_wmma_f32_16x16x4_f32(
        /*neg_a=*/false, a, /*neg_b=*/false, b,
        /*c_mod=*/(short)0, c, /*reuse_a=*/false, /*reuse_b=*/false);
}

// ---------------------------------------------------------------------------
// GEMM: C[M,N] = A[M,K] @ W[N,K]^T      (all dims multiples of tile sizes)
// 256 threads = 8 waves; block tile 128x128; K tile 32; double-buffered LDS
// with async global->LDS prefetch of tile k+1 overlapping compute of tile k.
// Wave (wm=w/2 in 0..3, wn=w&1): rows wm*32..+32 (2 frags), cols wn*64..+64 (4)
// ---------------------------------------------------------------------------
#define KP 36   // 32 + 4 pad: keeps 16B row alignment, conflict-free frag reads

__global__ __launch_bounds__(256) void gemm_xwt(
    const float* __restrict__ A, const float* __restrict__ W,
    float* __restrict__ C, int M, int N, int K)
{
    __shared__ float As[2][128][KP];
    __shared__ float Bs[2][128][KP];

    const int tid  = threadIdx.x;
    const int w    = tid >> 5;
    const int lane = tid & 31;
    const int lo   = lane & 15;
    const int hi   = lane >> 4;
    const int wm   = w >> 1;
    const int wn   = w & 1;
    const int m0   = blockIdx.y * 128;
    const int n0   = blockIdx.x * 128;

    const v8f vzero = {};
    v8f acc[2][4];
#pragma unroll
    for (int i = 0; i < 2; ++i)
#pragma unroll
        for (int j = 0; j < 4; ++j) acc[i][j] = vzero;

    // issue one tile's worth of async copies (128x32 of A and of W = 4 b128 each/thread)
    auto issue_tile = [&](int buf, int k0) {
#pragma unroll
        for (int t = 0; t < 4; ++t) {
            int fi  = tid + t * 256;
            int row = fi >> 3;
            int c4  = (fi & 7) << 2;
            async_cp_b128(&As[buf][row][c4], A + (size_t)(m0 + row) * K + k0 + c4);
            async_cp_b128(&Bs[buf][row][c4], W + (size_t)(n0 + row) * K + k0 + c4);
        }
    };

    const int nk = K >> 5;
    issue_tile(0, 0);

    for (int kt = 0; kt < nk; ++kt) {
        const int cur = kt & 1;
        async_wait0();          // my async copies for buf `cur` are in LDS
        __syncthreads();        // everyone's are; previous compute on buf cur^1 done
        if (kt + 1 < nk) issue_tile(cur ^ 1, (kt + 1) << 5);

#pragma unroll
        for (int kk = 0; kk < 32; kk += 4) {
            v2f a[2], b[4];
#pragma unroll
            for (int mi = 0; mi < 2; ++mi) {
                int m = wm * 32 + mi * 16 + lo;
                a[mi][0] = As[cur][m][kk + 2 * hi];
                a[mi][1] = As[cur][m][kk + 2 * hi + 1];
            }
#pragma unroll
            for (int ni = 0; ni < 4; ++ni) {
                int n = wn * 64 + ni * 16 + lo;
                b[ni][0] = Bs[cur][n][kk + 2 * hi];      // B[k][n] = W[n][k]
                b[ni][1] = Bs[cur][n][kk + 2 * hi + 1];
            }
#pragma unroll
            for (int mi = 0; mi < 2; ++mi)
#pragma unroll
                for (int ni = 0; ni < 4; ++ni)
                    acc[mi][ni] = wmma4(a[mi], b[ni], acc[mi][ni]);
        }
        __syncthreads();        // all reads of buf `cur` done before it is re-issued
    }

#pragma unroll
    for (int mi = 0; mi < 2; ++mi)
#pragma unroll
        for (int ni = 0; ni < 4; ++ni)
#pragma unroll
            for (int r = 0; r < 8; ++r) {
                int m = m0 + wm * 32 + mi * 16 + r + 8 * hi;
                int n = n0 + wn * 64 + ni * 16 + lo;
                C[(size_t)m * N + n] = acc[mi][ni][r];
            }
}

// ---------------------------------------------------------------------------
// RoPE + transpose: qkv_raw[B*T, 3*2048] -> Q,K,V each [B*H, T, 128]
// One thread per (b,t,h,d<64) rotary pair. 2*2048*16*64 = 4,194,304 threads.
// ---------------------------------------------------------------------------
__global__ __launch_bounds__(256) void rope_kernel(
    const float* __restrict__ qkv, const float* __restrict__ cosp,
    const float* __restrict__ sinp, float* __restrict__ Qb,
    float* __restrict__ Kb, float* __restrict__ Vb)
{
    int idx = blockIdx.x * 256 + threadIdx.x;
    int d = idx & 63;
    int h = (idx >> 6) & 15;
    int t = (idx >> 10) & 2047;
    int b = idx >> 21;

    size_t base = ((size_t)(b * 2048 + t)) * 6144 + h * 128 + d;
    float c = cosp[t * 128 + d];
    float s = sinp[t * 128 + d];

    float q1 = qkv[base],        q2 = qkv[base + 64];
    float k1 = qkv[base + 2048], k2 = qkv[base + 2048 + 64];
    float v1 = qkv[base + 4096], v2 = qkv[base + 4096 + 64];

    size_t ob = ((size_t)((b * 16 + h) * 2048 + t)) * 128 + d;
    Qb[ob] = q1 * c - q2 * s;  Qb[ob + 64] = q2 * c + q1 * s;
    Kb[ob] = k1 * c - k2 * s;  Kb[ob + 64] = k2 * c + k1 * s;
    Vb[ob] = v1;               Vb[ob + 64] = v2;
}

// ---------------------------------------------------------------------------
// Flash-style causal attention. Block = (qtile of 64 rows) x (one b,h).
// 256 threads = 8 waves. O tile 64x128: wave (wm=w/2) rows 16, (wn=w&1) cols 64.
// S tile 64x64: wave handles frags [wm][(wn*2)..(wn*2+1)].
// K/V tiles double-buffered; async prefetch of tile kt+1 overlaps S, softmax, PV.
// ---------------------------------------------------------------------------
#define QP 132  // 128 + 4 pad
#define SP 68   // 64 + 4 pad

__global__ __launch_bounds__(256) void attn_kernel(
    const float* __restrict__ Q, const float* __restrict__ K,
    const float* __restrict__ V, float* __restrict__ Oout)
{
    __shared__ float Qs[64][QP];
    __shared__ float Ks[2][64][QP];
    __shared__ float Vs[2][64][QP];
    __shared__ float Ps[64][SP];
    __shared__ float red[64][4];
    __shared__ float m_s[64], l_s[64], al_s[64];

    const int tid  = threadIdx.x;
    const int w    = tid >> 5;
    const int lane = tid & 31;
    const int lo   = lane & 15;
    const int hi   = lane >> 4;
    const int wm   = w >> 1;
    const int wn   = w & 1;
    const int qt   = blockIdx.x;            // q tile: 0..31
    const int bh   = blockIdx.y;            // 0..31
    const int q0   = qt * 64;
    const size_t headbase = (size_t)bh * 2048 * 128;
    const float scale = 0.08838834764831845f;  // 1/sqrt(128)

    auto issue_kv = [&](int buf, int kt) {
#pragma unroll
        for (int t = 0; t < 8; ++t) {
            int fi  = tid + t * 256;
            int row = fi >> 5;
            int c4  = (fi & 31) << 2;
            async_cp_b128(&Ks[buf][row][c4],
                          K + headbase + (size_t)(kt * 64 + row) * 128 + c4);
            async_cp_b128(&Vs[buf][row][c4],
                          V + headbase + (size_t)(kt * 64 + row) * 128 + c4);
        }
    };

    // Q tile (64x128) once + first K/V tile.
#pragma unroll
    for (int t = 0; t < 8; ++t) {
        int fi  = tid + t * 256;
        int row = fi >> 5;
        int c4  = (fi & 31) << 2;
        async_cp_b128(&Qs[row][c4],
                      Q + headbase + (size_t)(q0 + row) * 128 + c4);
    }
    issue_kv(0, 0);
    if (tid < 64) { m_s[tid] = -INFINITY; l_s[tid] = 0.0f; }

    const v8f vzero = {};
    v8f oacc[4];
#pragma unroll
    for (int i = 0; i < 4; ++i) oacc[i] = vzero;

    const int srow = tid >> 2;      // softmax: 4 threads per row
    const int part = tid & 3;

    for (int kt = 0; kt <= qt; ++kt) {
        const int cur = kt & 1;
        async_wait0();              // my copies for buf `cur` (and Q on kt==0) landed
        __syncthreads();            // everyone's landed; prior compute on cur^1 done
        if (kt < qt) issue_kv(cur ^ 1, kt + 1);

        // S = Q K^T  (per wave: 2 fragments of the 4x4 frag grid)
        v8f sacc[2];
        sacc[0] = vzero; sacc[1] = vzero;
#pragma unroll
        for (int kk = 0; kk < 128; kk += 4) {
            v2f a;
            int m = wm * 16 + lo;
            a[0] = Qs[m][kk + 2 * hi];
            a[1] = Qs[m][kk + 2 * hi + 1];
#pragma unroll
            for (int j = 0; j < 2; ++j) {
                int n = (wn * 2 + j) * 16 + lo;
                v2f b;
                b[0] = Ks[cur][n][kk + 2 * hi];       // (K^T)[k][n] = K[n][k]
                b[1] = Ks[cur][n][kk + 2 * hi + 1];
                sacc[j] = wmma4(a, b, sacc[j]);
            }
        }
#pragma unroll
        for (int j = 0; j < 2; ++j)
#pragma unroll
            for (int r = 0; r < 8; ++r)
                Ps[wm * 16 + r + 8 * hi][(wn * 2 + j) * 16 + lo] = sacc[j][r] * scale;
        __syncthreads();

        // --- online softmax on the 64x64 tile ---
        const int grow = q0 + srow;
        float lmax = -INFINITY;
#pragma unroll
        for (int c = 0; c < 16; ++c) {
            int col  = part * 16 + c;
            int gcol = kt * 64 + col;
            float v  = Ps[srow][col];
            if (gcol > grow) { v = -INFINITY; Ps[srow][col] = v; }
            lmax = fmaxf(lmax, v);
        }
        red[srow][part] = lmax;
        __syncthreads();
        if (part == 0) {
            float tm = fmaxf(fmaxf(red[srow][0], red[srow][1]),
                             fmaxf(red[srow][2], red[srow][3]));
            float old_m = m_s[srow];
            float new_m = fmaxf(old_m, tm);       // finite: col 0 always unmasked
            al_s[srow]  = __expf(old_m - new_m);  // -inf old -> alpha 0
            m_s[srow]   = new_m;
        }
        __syncthreads();
        float new_m = m_s[srow];
        float lsum = 0.0f;
#pragma unroll
        for (int c = 0; c < 16; ++c) {
            int col = part * 16 + c;
            float p = __expf(Ps[srow][col] - new_m);
            Ps[srow][col] = p;
            lsum += p;
        }
        red[srow][part] = lsum;
        __syncthreads();
        if (part == 0) {
            float s = red[srow][0] + red[srow][1] + red[srow][2] + red[srow][3];
            l_s[srow] = l_s[srow] * al_s[srow] + s;
        }
        __syncthreads();

        // Rescale O by alpha, then O += P @ V
#pragma unroll
        for (int ni = 0; ni < 4; ++ni)
#pragma unroll
            for (int r = 0; r < 8; ++r)
                oacc[ni][r] *= al_s[wm * 16 + r + 8 * hi];

#pragma unroll
        for (int kk = 0; kk < 64; kk += 4) {
            v2f a;
            int m = wm * 16 + lo;
            a[0] = Ps[m][kk + 2 * hi];
            a[1] = Ps[m][kk + 2 * hi + 1];
#pragma unroll
            for (int ni = 0; ni < 4; ++ni) {
                int n = wn * 64 + ni * 16 + lo;
                v2f b;
                b[0] = Vs[cur][kk + 2 * hi][n];
                b[1] = Vs[cur][kk + 2 * hi + 1][n];
                oacc[ni] = wmma4(a, b, oacc[ni]);
            }
        }
        // no trailing barrier: next iteration's wait+barrier protects buffer reuse
    }

    // Normalize and write out as [B, T, H*128] (contiguous D for the out-proj GEMM)
    int b = bh >> 4, h = bh & 15;
#pragma unroll
    for (int ni = 0; ni < 4; ++ni)
#pragma unroll
        for (int r = 0; r < 8; ++r) {
            int rl = wm * 16 + r + 8 * hi;
            int t  = q0 + rl;
            int d  = wn * 64 + ni * 16 + lo;
            float inv = 1.0f / l_s[rl];
            Oout[((size_t)(b * 2048 + t)) * 2048 + h * 128 + d] = oacc[ni][r] * inv;
        }
}

// ---------------------------------------------------------------------------
// Host side
// ---------------------------------------------------------------------------
extern "C" void kernel_launch(void* const* d_in, const int* in_sizes, int n_in,
                              void* d_out, int out_size, void* d_ws, size_t ws_size,
                              hipStream_t stream)
{
    const float* x    = (const float*)d_in[0];  // (2,2048,2048)
    const float* cosp = (const float*)d_in[1];  // (1,2048,128)
    const float* sinp = (const float*)d_in[2];  // (1,2048,128)
    const float* Wqkv = (const float*)d_in[3];  // (6144,2048)
    const float* Wout = (const float*)d_in[4];  // (2048,2048)
    float* out = (float*)d_out;                 // (2,2048,2048)

    float* ws      = (float*)d_ws;
    float* qkv_raw = ws;                                // 4096*6144
    float* Qb      = ws + (size_t)4096 * 6144;          // 32*2048*128
    float* Kb      = Qb + (size_t)32 * 2048 * 128;
    float* Vb      = Kb + (size_t)32 * 2048 * 128;
    float* attn    = qkv_raw;                           // reuse: qkv dead after RoPE

    // 1) qkv = x @ Wqkv^T   [4096,2048] x [6144,2048]^T
    dim3 g1(6144 / 128, 4096 / 128);
    gemm_xwt<<<g1, 256, 0, stream>>>(x, Wqkv, qkv_raw, 4096, 6144, 2048);

    // 2) RoPE + head transpose
    rope_kernel<<<(2 * 2048 * 16 * 64) / 256, 256, 0, stream>>>(
        qkv_raw, cosp, sinp, Qb, Kb, Vb);

    // 3) causal flash attention
    dim3 g2(2048 / 64, 32);
    attn_kernel<<<g2, 256, 0, stream>>>(Qb, Kb, Vb, attn);

    // 4) out = attn @ Wout^T
    dim3 g3(2048 / 128, 4096 / 128);
    gemm_xwt<<<g3, 256, 0, stream>>>(attn, Wout, out, 4096, 2048, 2048);
}